// HashedInterpolator_54314156425376
// MI455X (gfx1250) — compile-verified
//
#include <hip/hip_runtime.h>
#include <stdint.h>

// InstantNGP-style hash-grid interpolation, tuned for MI455X (gfx1250):
//  - gather-latency bound; 64MB table is L2-resident (192MB L2)
//  - per-lane async gathers (global_load_async_to_lds_b128, ASYNCcnt) keep
//    VGPR pressure minimal -> max occupancy for latency hiding
//  - split s_wait_asynccnt: overlap accumulation of the first 4 corners with
//    the in-flight gathers of the last 4 (async loads complete in order)
//  - non-temporal output stores protect L2 residency of the table

typedef float v4f __attribute__((ext_vector_type(4)));

#define BLOCK 128
#define HASH_MASK 0x1FFFFFu              // n_entries = 2^21
#define P1 19349663u
#define P2 83492791u
#define SLOT 48u                          // bytes per (thread,corner) stage slot (16B aligned, bank-conflict-free)
#define CORNER_STRIDE (BLOCK * SLOT)      // 6144 B
#define STAGE_BYTES (8u * CORNER_STRIDE)  // 49152 B dynamic LDS

extern __shared__ char hi_stage[];

__global__ __launch_bounds__(BLOCK) void hashed_interp_kernel(
    const float* __restrict__ pos,
    const float* __restrict__ table,
    const float* __restrict__ grids,
    float* __restrict__ out,
    int batch)
{
    const int tid = threadIdx.x;
    const int b = blockIdx.x * BLOCK + tid;
    if (b >= batch) return;

    const float x = pos[3 * b + 0];
    const float y = pos[3 * b + 1];
    const float z = pos[3 * b + 2];

    const int ix = (int)floorf(x * grids[0]);
    const int iy = (int)floorf(y * grids[1]);
    const int iz = (int)floorf(z * grids[2]);

    // hash terms: per-step floor-mod by 2^21 collapses exactly to a final mask
    // (power-of-two divisor; & distributes over ^; int32 wraparound multiply).
    const uint32_t cx0 = (uint32_t)ix,       cx1 = cx0 + 1u;
    const uint32_t ty0 = (uint32_t)iy * P1,  ty1 = ty0 + P1;
    const uint32_t tz0 = (uint32_t)iz * P2,  tz1 = tz0 + P2;

    const uint64_t tbase = (uint64_t)(uintptr_t)table;
    const uint32_t lds0  = (uint32_t)tid * SLOT;

    // Issue all 16 async gathers (8 corners x 32B). No destination VGPRs;
    // tracked with ASYNCcnt. offset:16 applies to both global and LDS sides.
#pragma unroll
    for (int c = 0; c < 8; ++c) {
        const uint32_t h = (((c & 4) ? cx1 : cx0) ^
                            ((c & 2) ? ty1 : ty0) ^
                            ((c & 1) ? tz1 : tz0)) & HASH_MASK;
        const uint64_t ga  = tbase + (uint64_t)h * 32u;   // 8 features * 4B
        const uint32_t lds = lds0 + (uint32_t)c * CORNER_STRIDE;
        asm volatile(
            "global_load_async_to_lds_b128 %0, %1, off\n\t"
            "global_load_async_to_lds_b128 %0, %1, off offset:16"
            :: "v"(lds), "v"(ga) : "memory");
    }

    // Interpolation weights while the gathers are in flight.
    // Faithful to reference: uses position itself (bit==0 -> p, bit==1 -> 1-p),
    // corner order = itertools.product bits (x=bit2, y=bit1, z=bit0).
    const float fx1 = 1.0f - x, fy1 = 1.0f - y, fz1 = 1.0f - z;
    float w[8];
#pragma unroll
    for (int c = 0; c < 8; ++c)
        w[c] = ((c & 4) ? fx1 : x) * ((c & 2) ? fy1 : y) * ((c & 1) ? fz1 : z);

    v4f accLo = {0.0f, 0.0f, 0.0f, 0.0f};
    v4f accHi = {0.0f, 0.0f, 0.0f, 0.0f};

    // Async loads complete in order: asynccnt <= 8 means corners 0..3 landed.
    asm volatile("s_wait_asynccnt 8" ::: "memory");
#pragma unroll
    for (int c = 0; c < 4; ++c) {
        const v4f* p = (const v4f*)(hi_stage + lds0 + (uint32_t)c * CORNER_STRIDE);
        const v4f lo = p[0];
        const v4f hi = p[1];
        accLo += lo * w[c];
        accHi += hi * w[c];
    }

    asm volatile("s_wait_asynccnt 0" ::: "memory");
#pragma unroll
    for (int c = 4; c < 8; ++c) {
        const v4f* p = (const v4f*)(hi_stage + lds0 + (uint32_t)c * CORNER_STRIDE);
        const v4f lo = p[0];
        const v4f hi = p[1];
        accLo += lo * w[c];
        accHi += hi * w[c];
    }

    // Non-temporal stores: don't let the 32MB output evict the L2-resident table.
    v4f* o = (v4f*)(out + (size_t)b * 8u);
    __builtin_nontemporal_store(accLo, o + 0);
    __builtin_nontemporal_store(accHi, o + 1);
}

extern "C" void kernel_launch(void* const* d_in, const int* in_sizes, int n_in,
                              void* d_out, int out_size, void* d_ws, size_t ws_size,
                              hipStream_t stream) {
    const float* pos   = (const float*)d_in[0];
    const float* table = (const float*)d_in[1];
    const float* grids = (const float*)d_in[2];
    float* out = (float*)d_out;

    const int batch = in_sizes[0] / 3;
    const int nblk  = (batch + BLOCK - 1) / BLOCK;

    hipLaunchKernelGGL(hashed_interp_kernel, dim3(nblk), dim3(BLOCK),
                       STAGE_BYTES, stream, pos, table, grids, out, batch);
}